// VQEmbedding_56925496541800
// MI455X (gfx1250) — compile-verified
//
#include <hip/hip_runtime.h>

// ---------------- problem constants ----------------
#define KCODES 8192
#define DDIM   256
#define BB     16
#define HH     32
#define WW     32
#define NVEC   (BB * HH * WW)          // 16384 latent vectors
#define EPSF   1e-5f
#define DECAYF 0.99f

typedef __attribute__((ext_vector_type(2))) float v2f;
typedef __attribute__((ext_vector_type(8))) float v8f;

// ---------------- workspace layout (in floats) ----------------
#define OFF_WNORM 0
#define OFF_IDX   (OFF_WNORM + KCODES)
#define OFF_SIZE  (OFF_IDX + NVEC)
#define OFF_SSUM  (OFF_SIZE + KCODES)
#define OFF_N     (OFF_SSUM + KCODES * DDIM)
#define OFF_NW    (OFF_N + 1)

// ---------------- kernel 1: codebook norms ----------------
__global__ __launch_bounds__(256) void wnorm_kernel(const float* __restrict__ w,
                                                    float* __restrict__ wnorm) {
    int k = blockIdx.x * blockDim.x + threadIdx.x;
    if (k >= KCODES) return;
    const float4* row = (const float4*)(w + (size_t)k * DDIM);
    float s = 0.f;
#pragma unroll 8
    for (int i = 0; i < DDIM / 4; ++i) {
        float4 v = row[i];
        s += v.x * v.x + v.y * v.y + v.z * v.z + v.w * v.w;
    }
    wnorm[k] = s;
}

// ---------------- kernel 2: distances via f32 WMMA + argmin ----------------
// One workgroup (8 waves) per 32 z-rows (two 16-row A tiles in LDS).
// Each wave owns a disjoint set of 16-code column tiles; each B fragment
// load feeds TWO v_wmma_f32_16x16x4_f32 (two independent accumulators).
// score(m, n) = ||w_n||^2 - 2 * dot(z_m, w_n)   (||z||^2 dropped: constant/row)
__global__ __launch_bounds__(256) void dist_argmin_kernel(
    const float* __restrict__ z_e_x,    // [B, D, H, W]
    const float* __restrict__ weight,   // [K, D]
    const float* __restrict__ wnorm,    // [K]
    int* __restrict__ indices)          // [N]
{
    constexpr int ZPAD = DDIM + 4;      // 260-float row stride: conflict-free b64 reads
    __shared__ float zt[32 * ZPAD];     // ~33 KB
    __shared__ float sb_s[8][32];
    __shared__ int   sb_i[8][32];

    const int tid  = threadIdx.x;
    const int wave = tid >> 5;
    const int lane = tid & 31;
    const int hf   = lane >> 4;         // K-split half of the wave
    const int l16  = lane & 15;
    const int row_base = blockIdx.x * 32;

    // ---- stage 32 x 256 z tile (gather NHWC rows out of NCHW input) ----
    for (int e = tid; e < 32 * DDIM; e += 256) {
        const int r = e & 31;           // row within tile
        const int d = e >> 5;
        const int gn = row_base + r;
        const int b = gn >> 10, rem = gn & 1023, h = rem >> 5, ww = rem & 31;
        zt[r * ZPAD + d] =
            z_e_x[(size_t)b * (DDIM * HH * WW) + (size_t)d * (HH * WW) + h * WW + ww];
    }
    __syncthreads();

    float best_s0[8], best_s1[8];
    int   best_i0[8], best_i1[8];
#pragma unroll
    for (int r = 0; r < 8; ++r) {
        best_s0[r] = 3.4e38f; best_i0[r] = 0;
        best_s1[r] = 3.4e38f; best_i1[r] = 0;
    }

    // A fragment sources: tile0 row = l16, tile1 row = 16 + l16; K-pair = 2*hf
    const float* aptr0 = &zt[l16 * ZPAD + 2 * hf];
    const float* aptr1 = &zt[(16 + l16) * ZPAD + 2 * hf];

    // 512 column tiles uniformly split over 8 waves -> EXEC all-ones for WMMA
    for (int jt = wave; jt < KCODES / 16; jt += 8) {
        const int col = jt * 16 + l16;                       // this lane's codebook row
        const float* bptr = weight + (size_t)col * DDIM + 2 * hf;
        __builtin_prefetch(bptr + 128 * DDIM, 0, 0);         // next tile for this wave

        v8f c0 = {0.f, 0.f, 0.f, 0.f, 0.f, 0.f, 0.f, 0.f};
        v8f c1 = {0.f, 0.f, 0.f, 0.f, 0.f, 0.f, 0.f, 0.f};
#pragma unroll 8
        for (int ks = 0; ks < DDIM / 4; ++ks) {
            v2f b  = *(const v2f*)(bptr  + ks * 4);          // global_load_b64 (shared)
            v2f a0 = *(const v2f*)(aptr0 + ks * 4);          // ds_load_b64
            v2f a1 = *(const v2f*)(aptr1 + ks * 4);          // ds_load_b64
            c0 = __builtin_amdgcn_wmma_f32_16x16x4_f32(
                     false, a0, false, b, (short)0, c0, false, false);
            c1 = __builtin_amdgcn_wmma_f32_16x16x4_f32(
                     false, a1, false, b, (short)0, c1, false, false);
        }

        // C/D layout: lane holds column n = col; rows m = r + 8*hf (+16 for tile1)
        const float wn = wnorm[col];
#pragma unroll
        for (int r = 0; r < 8; ++r) {
            const float s0 = wn - 2.0f * c0[r];
            if (s0 < best_s0[r] || (s0 == best_s0[r] && col < best_i0[r])) {
                best_s0[r] = s0; best_i0[r] = col;
            }
            const float s1 = wn - 2.0f * c1[r];
            if (s1 < best_s1[r] || (s1 == best_s1[r] && col < best_i1[r])) {
                best_s1[r] = s1; best_i1[r] = col;
            }
        }
    }

    // cross-lane argmin within each 16-lane half (same rows, different columns)
#pragma unroll
    for (int r = 0; r < 8; ++r) {
        float s0 = best_s0[r]; int i0 = best_i0[r];
        float s1 = best_s1[r]; int i1 = best_i1[r];
#pragma unroll
        for (int off = 8; off > 0; off >>= 1) {
            float os = __shfl_xor(s0, off, 16);
            int   oi = __shfl_xor(i0, off, 16);
            if (os < s0 || (os == s0 && oi < i0)) { s0 = os; i0 = oi; }
            os = __shfl_xor(s1, off, 16);
            oi = __shfl_xor(i1, off, 16);
            if (os < s1 || (os == s1 && oi < i1)) { s1 = os; i1 = oi; }
        }
        if (l16 == 0) {
            sb_s[wave][hf * 8 + r]      = s0; sb_i[wave][hf * 8 + r]      = i0;
            sb_s[wave][16 + hf * 8 + r] = s1; sb_i[wave][16 + hf * 8 + r] = i1;
        }
    }
    __syncthreads();

    // combine the 8 waves' disjoint column ranges
    if (tid < 32) {
        float s  = sb_s[0][tid];
        int   bi = sb_i[0][tid];
#pragma unroll
        for (int wv = 1; wv < 8; ++wv) {
            float os = sb_s[wv][tid];
            int   oi = sb_i[wv][tid];
            if (os < s || (os == s && oi < bi)) { s = os; bi = oi; }
        }
        indices[row_base + tid] = bi;
    }
}

// ---------------- kernel 3: zero scratch ----------------
__global__ __launch_bounds__(256) void zero_kernel(float* __restrict__ p, int n) {
    int i = blockIdx.x * blockDim.x + threadIdx.x;
    if (i < n) p[i] = 0.f;
}

// ---------------- kernel 4: segment sums (counts + sum of z) ----------------
__global__ __launch_bounds__(256) void scatter_kernel(const float* __restrict__ z_e_x,
                                                      const int* __restrict__ indices,
                                                      float* __restrict__ size,
                                                      float* __restrict__ ssum) {
    const int n = blockIdx.x;
    const int d = threadIdx.x;
    const int idx = indices[n];
    const int b = n >> 10, rem = n & 1023, h = rem >> 5, ww = rem & 31;
    const float v = z_e_x[(size_t)b * (DDIM * HH * WW) + (size_t)d * (HH * WW) + h * WW + ww];
    atomicAdd(&ssum[(size_t)idx * DDIM + d], v);
    if (d == 0) atomicAdd(&size[idx], 1.0f);
}

// ---------------- kernel 5: n = sum(rs_new) ----------------
__global__ __launch_bounds__(256) void reduce_n_kernel(const float* __restrict__ running_size,
                                                       const float* __restrict__ size,
                                                       float* __restrict__ nout) {
    __shared__ float red[256];
    float s = 0.f;
    for (int k = threadIdx.x; k < KCODES; k += 256)
        s += DECAYF * running_size[k] + (1.0f - DECAYF) * size[k];
    red[threadIdx.x] = s;
    __syncthreads();
    for (int st = 128; st > 0; st >>= 1) {
        if (threadIdx.x < st) red[threadIdx.x] += red[threadIdx.x + st];
        __syncthreads();
    }
    if (threadIdx.x == 0) nout[0] = red[0];
}

// ---------------- kernel 6: EMA codebook update ----------------
__global__ __launch_bounds__(256) void newweight_kernel(const float* __restrict__ running_size,
                                                        const float* __restrict__ running_sum,
                                                        const float* __restrict__ size,
                                                        const float* __restrict__ ssum,
                                                        const float* __restrict__ nptr,
                                                        float* __restrict__ nw) {
    const int k = blockIdx.x, d = threadIdx.x;
    const float n  = nptr[0];
    const float rs = DECAYF * running_size[k] + (1.0f - DECAYF) * size[k];
    const float dn = (rs + EPSF) / (n + (float)KCODES * EPSF) * n;
    const float rsum = DECAYF * running_sum[(size_t)k * DDIM + d]
                     + (1.0f - DECAYF) * ssum[(size_t)k * DDIM + d];
    nw[(size_t)k * DDIM + d] = rsum / dn;
}

// ---------------- kernel 7: double gather into NCHW outputs ----------------
__global__ __launch_bounds__(256) void gather_kernel(const int* __restrict__ indices,
                                                     const float* __restrict__ weight,
                                                     const float* __restrict__ nw,
                                                     float* __restrict__ out) {
    const int n = blockIdx.x, d = threadIdx.x;
    const int idx = indices[n];
    const int b = n >> 10, rem = n & 1023, h = rem >> 5, ww = rem & 31;
    const size_t o = (size_t)b * (DDIM * HH * WW) + (size_t)d * (HH * WW) + h * WW + ww;
    out[o] = weight[(size_t)idx * DDIM + d];                          // z_q_x
    out[(size_t)NVEC * DDIM + o] = nw[(size_t)idx * DDIM + d];        // z_q_x_bar
}

// ---------------- launch ----------------
extern "C" void kernel_launch(void* const* d_in, const int* in_sizes, int n_in,
                              void* d_out, int out_size, void* d_ws, size_t ws_size,
                              hipStream_t stream) {
    (void)in_sizes; (void)n_in; (void)out_size; (void)ws_size;
    const float* z_e_x        = (const float*)d_in[0];
    const float* weight       = (const float*)d_in[1];
    const float* running_size = (const float*)d_in[2];
    const float* running_sum  = (const float*)d_in[3];

    float* ws      = (float*)d_ws;
    float* wnorm   = ws + OFF_WNORM;
    int*   indices = (int*)(ws + OFF_IDX);
    float* size    = ws + OFF_SIZE;
    float* ssum    = ws + OFF_SSUM;    // contiguous after size
    float* nptr    = ws + OFF_N;
    float* nw      = ws + OFF_NW;

    wnorm_kernel<<<KCODES / 256, 256, 0, stream>>>(weight, wnorm);
    zero_kernel<<<(KCODES + KCODES * DDIM + 255) / 256, 256, 0, stream>>>(
        size, KCODES + KCODES * DDIM);
    dist_argmin_kernel<<<NVEC / 32, 256, 0, stream>>>(z_e_x, weight, wnorm, indices);
    scatter_kernel<<<NVEC, 256, 0, stream>>>(z_e_x, indices, size, ssum);
    reduce_n_kernel<<<1, 256, 0, stream>>>(running_size, size, nptr);
    newweight_kernel<<<KCODES, DDIM, 0, stream>>>(running_size, running_sum,
                                                  size, ssum, nptr, nw);
    gather_kernel<<<NVEC, 256, 0, stream>>>(indices, weight, nw, (float*)d_out);
}